// Conv4d_16441134809547
// MI455X (gfx1250) — compile-verified
//
#include <hip/hip_runtime.h>

typedef float v2f __attribute__((ext_vector_type(2)));
typedef float v8f __attribute__((ext_vector_type(8)));

// ---- problem constants (from the reference) ----
#define B_   4
#define T_   20
#define Z_   20
#define H_   40
#define W_   40
#define CI   8
#define CO   16
#define TO   18
#define ZO   18
#define HO   38
#define WO   38
#define KTOT 648                      // 3*3*3*3*8
#define MTOT 1871424L                 // 4*18*18*38*38

// x strides in floats (channels-last)
#define ST_  128000                   // Z_*H_*W_*CI
#define SZ_  12800                    // H_*W_*CI
#define SH_  320                      // W_*CI

__global__ __launch_bounds__(256)
void conv4d_wmma_f32(const float* __restrict__ x,
                     const float* __restrict__ wgt,
                     const float* __restrict__ bias,
                     float* __restrict__ out) {
    // Full flattened weight matrix B[K=648][N=16] in LDS (41.5 KB), b128 staged.
    __shared__ __align__(16) float lds_w[KTOT * CO];

    const int tid = threadIdx.x;
    for (int i = tid; i < (KTOT * CO) / 4; i += 256)
        ((float4*)lds_w)[i] = ((const float4*)wgt)[i];
    __syncthreads();

    const int lane = tid & 31;
    const int wave = tid >> 5;
    const int n    = lane & 15;       // Cout column this lane holds for B/C/D
    const int hi   = lane >> 4;       // 0: rows 0-7, K {0,1}; 1: rows 8-15, K {2,3}
    const int koff = hi << 1;         // per-lane K sub-offset within a 4-step

    const long mBase = (long)blockIdx.x * 128 + (long)wave * 16;

    // Per-lane A row (output position), clamped so OOB lanes read safe memory.
    long mA = mBase + n;
    if (mA >= MTOT) mA = MTOT - 1;
    int r = (int)mA;
    const int w = r % WO; r /= WO;
    const int h = r % HO; r /= HO;
    const int z = r % ZO; r /= ZO;
    const int t = r % TO;
    const int b = r / TO;
    const long base = (((((long)b * T_ + t) * Z_ + z) * H_ + h) * W_ + w) * CI;
    // Fold the per-lane K sub-offset into the A pointer once.
    const float* __restrict__ xb = x + base + koff;

    const float bv = bias[n];         // hoisted: overlaps with main loop

    // Two independent accumulation chains (step parity) -> consecutive WMMAs
    // are not serialized on C, and DS waits can retire progressively.
    v8f acc0 = {};
    v8f acc1 = {};

    // K = (dt,dz,dh,dw,c): one (dt,dz) slab = 72 K = 18 WMMA steps.
    for (int dt = 0; dt < 3; ++dt) {
        for (int dz = 0; dz < 3; ++dz) {
            const float* __restrict__ p0 = xb + (long)dt * ST_ + dz * SZ_;
            const int kslab = (dt * 3 + dz) * 72;

            // ---- phase 1: issue ALL 18 A-loads and 18 B-loads (registers) ----
            v2f Af[18];
            v2f Bf[18];
            #pragma unroll
            for (int s = 0; s < 18; ++s) {
                const int dh   = s / 6;
                const int dw   = (s >> 1) % 3;
                const int half = s & 1;
                const float2 av =
                    *(const float2*)(p0 + dh * SH_ + dw * CI + half * 4);
                Af[s][0] = av.x; Af[s][1] = av.y;

                const int k0 = kslab + s * 4 + koff;
                Bf[s][0] = lds_w[k0 * CO + n];
                Bf[s][1] = lds_w[(k0 + 1) * CO + n];
            }

            // ---- phase 2: 18 WMMAs, alternating accumulator chains ----
            #pragma unroll
            for (int s = 0; s < 18; ++s) {
                if (s & 1)
                    acc1 = __builtin_amdgcn_wmma_f32_16x16x4_f32(
                        false, Af[s], false, Bf[s], (short)0, acc1, false, false);
                else
                    acc0 = __builtin_amdgcn_wmma_f32_16x16x4_f32(
                        false, Af[s], false, Bf[s], (short)0, acc0, false, false);
            }
        }
    }

    v8f acc = acc0 + acc1;

    // Epilogue: bias add + store of the 8 accumulator rows this lane holds.
    if (mBase + 16 <= MTOT) {
        // Fast path: full tile, no guards.
        #pragma unroll
        for (int i = 0; i < 8; ++i) {
            const long m = mBase + (hi ? 8 : 0) + i;
            out[m * CO + n] = acc[i] + bv;
        }
    } else {
        #pragma unroll
        for (int i = 0; i < 8; ++i) {
            const long m = mBase + (hi ? 8 : 0) + i;
            if (m < MTOT)
                out[m * CO + n] = acc[i] + bv;
        }
    }
}

extern "C" void kernel_launch(void* const* d_in, const int* in_sizes, int n_in,
                              void* d_out, int out_size, void* d_ws, size_t ws_size,
                              hipStream_t stream) {
    const float* x    = (const float*)d_in[0];   // [4,20,20,40,40,8]
    const float* wgt  = (const float*)d_in[1];   // [3,3,3,3,8,16] == B[648][16]
    const float* bias = (const float*)d_in[2];   // [16]
    float*       out  = (float*)d_out;           // [4,18,18,38,38,16]

    const long blocks = (MTOT + 127) / 128;      // 128 output positions per block
    conv4d_wmma_f32<<<(int)blocks, 256, 0, stream>>>(x, wgt, bias, out);
}